// H2T_3633542332964
// MI455X (gfx1250) — compile-verified
//
#include <hip/hip_runtime.h>

typedef float v2f __attribute__((ext_vector_type(2)));
typedef float v8f __attribute__((ext_vector_type(8)));

#define DIMS 768
#define NPROTO 16

// ---------------------------------------------------------------- zero scratch
__global__ void zero_f32(float* __restrict__ a, int n) {
    int i = blockIdx.x * blockDim.x + threadIdx.x;
    if (i < n) a[i] = 0.0f;
}

// ------------------------------------------------- row-normalize prototypes
__global__ __launch_bounds__(256) void norm_protos(const float* __restrict__ protos,
                                                   float* __restrict__ pn) {
    __shared__ float red[8];
    const int b = blockIdx.x;
    const float* row = protos + (size_t)b * DIMS;
    float ss = 0.0f;
    for (int d = threadIdx.x; d < DIMS; d += 256) {
        float v = row[d];
        ss = fmaf(v, v, ss);
    }
    for (int off = 16; off > 0; off >>= 1) ss += __shfl_xor(ss, off, 32);
    if ((threadIdx.x & 31) == 0) red[threadIdx.x >> 5] = ss;
    __syncthreads();
    if (threadIdx.x == 0) {
        float s = 0.0f;
        for (int w = 0; w < 8; ++w) s += red[w];
        red[0] = s;
    }
    __syncthreads();
    const float inv = 1.0f / sqrtf(red[0]);
    for (int d = threadIdx.x; d < DIMS; d += 256) pn[(size_t)b * DIMS + d] = row[d] * inv;
}

// IEEE-754 order-preserving map: unsigned compare of ord(a) == float compare of a
__device__ __forceinline__ unsigned int f32_ordered(float f) {
    unsigned int b = __float_as_uint(f);
    return b ^ (((int)b >> 31) | 0x80000000u);
}

// --------------------------------------- fused scoring + argmax + scatter-sum
// One wave == one 16-row tile. WMMA f32 16x16x4: A = 16 rows x 4 dims,
// B = 4 dims x 16 prototypes, accumulated over K = 768.
__global__ __launch_bounds__(256) void assign_pool(const float* __restrict__ x,
                                                   const float* __restrict__ pn,
                                                   float* __restrict__ gsums,
                                                   float* __restrict__ gcnt,
                                                   int ntiles) {
    __shared__ float ldsSums[NPROTO * DIMS];   // 48 KB block-private partial sums
    __shared__ float ldsCnt[NPROTO];
    for (int i = threadIdx.x; i < NPROTO * DIMS; i += 256) ldsSums[i] = 0.0f;
    if (threadIdx.x < NPROTO) ldsCnt[threadIdx.x] = 0.0f;
    __syncthreads();

    const int wave  = threadIdx.x >> 5;
    const int lane  = threadIdx.x & 31;
    const int laneM = lane & 15;          // row (for A) / prototype column (for B)
    const int kOff  = (lane >> 4) << 1;   // K,K+1 in lanes 0-15; K+2,K+3 in 16-31
    const float* brow = pn + (size_t)laneM * DIMS + kOff;

    for (int t = blockIdx.x * 8 + wave; t < ntiles; t += gridDim.x * 8) {
        const float* tile = x + (size_t)t * 16 * DIMS;
        const float* arow = tile + (size_t)laneM * DIMS + kOff;

        v8f acc = {0.f, 0.f, 0.f, 0.f, 0.f, 0.f, 0.f, 0.f};
        float ss = 0.0f;
        #pragma unroll 8
        for (int k = 0; k < DIMS; k += 4) {
            v2f a = *(const v2f*)(arow + k);
            v2f b = *(const v2f*)(brow + k);
            acc = __builtin_amdgcn_wmma_f32_16x16x4_f32(
                false, a, false, b, (short)0, acc, false, false);
            ss = fmaf(a.x, a.x, fmaf(a.y, a.y, ss));
        }
        // combine the two K-halves: lane L now holds sum(x[row L&15]^2)
        ss += __shfl_xor(ss, 16, 32);

        // D layout: lane = proto (within 16-lane half), acc[j] = row (j + 8*half).
        // Branchless argmax across the 16 lanes of each half using a sortable
        // 64-bit key: [ ordered(sim) : 32 | (15 - proto) : 32 ].  Max key ==
        // max sim, exact f32 tie -> smaller proto index (matches jnp.argmin).
        int assign[8];
        #pragma unroll
        for (int j = 0; j < 8; ++j) {
            unsigned long long key =
                ((unsigned long long)f32_ordered(acc[j]) << 32) |
                (unsigned int)(15 - laneM);
            #pragma unroll
            for (int off = 1; off < 16; off <<= 1) {
                unsigned long long ok = __shfl_xor(key, off, 32);
                key = (ok > key) ? ok : key;
            }
            assign[j] = 15 - (int)(key & 0xFu);   // uniform across the half
        }

        // Scatter-accumulate normalized rows into LDS partial sums.
        // Rows were just streamed -> second read is WGP$/L2 resident.
        #pragma unroll
        for (int r = 0; r < 16; ++r) {
            const int   p    = __shfl(assign[r & 7], (r >> 3) << 4, 32);
            const float inv  = 1.0f / sqrtf(__shfl(ss, r, 32));
            const float* xr  = tile + r * DIMS;
            float*       dst = ldsSums + p * DIMS;
            for (int d = lane; d < DIMS; d += 32)
                atomicAdd(dst + d, xr[d] * inv);     // ds_add_f32, conflict-free banks
            if (lane == 0) atomicAdd(&ldsCnt[p], 1.0f);
        }
    }

    __syncthreads();
    for (int i = threadIdx.x; i < NPROTO * DIMS; i += 256)
        atomicAdd(&gsums[i], ldsSums[i]);
    if (threadIdx.x < NPROTO) atomicAdd(&gcnt[threadIdx.x], ldsCnt[threadIdx.x]);
}

// ---------------------------------------------------------------- means
__global__ void finalize_means(const float* __restrict__ gsums,
                               const float* __restrict__ gcnt,
                               float* __restrict__ out) {
    int i = blockIdx.x * blockDim.x + threadIdx.x;
    if (i < NPROTO * DIMS) {
        float c = gcnt[i / DIMS];
        out[i] = (c > 0.0f) ? gsums[i] / fmaxf(c, 1.0f) : 0.0f;
    }
}

extern "C" void kernel_launch(void* const* d_in, const int* in_sizes, int n_in,
                              void* d_out, int out_size, void* d_ws, size_t ws_size,
                              hipStream_t stream) {
    const float* x      = (const float*)d_in[0];   // [1, N, 768] f32
    const float* protos = (const float*)d_in[1];   // [16, 768]   f32
    float* out = (float*)d_out;                    // [1, 16*768] f32
    float* ws  = (float*)d_ws;

    float* pn    = ws;                         // 16*768 normalized prototypes
    float* gsums = ws + NPROTO * DIMS;         // 16*768 accumulators
    float* gcnt  = ws + 2 * NPROTO * DIMS;     // 16 counts

    const int N      = in_sizes[0] / DIMS;     // 200000 (multiple of 16)
    const int ntiles = N / 16;

    const int accn = NPROTO * DIMS + NPROTO;
    hipLaunchKernelGGL(zero_f32, dim3((accn + 255) / 256), dim3(256), 0, stream,
                       gsums, accn);
    hipLaunchKernelGGL(norm_protos, dim3(NPROTO), dim3(256), 0, stream, protos, pn);
    hipLaunchKernelGGL(assign_pool, dim3(512), dim3(256), 0, stream,
                       x, pn, gsums, gcnt, ntiles);
    hipLaunchKernelGGL(finalize_means, dim3((NPROTO * DIMS + 255) / 256), dim3(256),
                       0, stream, gsums, gcnt, out);
}